// _Renderer_2851858284921
// MI455X (gfx1250) — compile-verified
//
#include <hip/hip_runtime.h>
#include <hip/hip_bf16.h>

typedef __attribute__((ext_vector_type(2))) float v2f;
typedef __attribute__((ext_vector_type(8))) float v8f;

#define BATCH 2
#define NVERT 1282
#define NFACE 1280          // faces per batch in input
#define F2    2560          // faces after winding duplication
#define HRES  256           // supersampled raster resolution
#define ORES  128           // output resolution
#define NEARZ 0.1f
#define FARZ  100.0f
#define EPSD  0.001f
#define EYE_OFF 2.732050807568877f   // -EYE_Z = 1/tan(30deg)+1
#define PW      0.5773502691896258f  // tan(30deg)

__device__ __forceinline__ float axf(int i) {
    // (2i + 1 - H) / H
    return (2.0f * (float)i + 1.0f - (float)HRES) * (1.0f / (float)HRES);
}

// ---------------------------------------------------------------------------
// Kernel 1: per-face preprocessing.
// Produces, for each of B*F2 faces:
//   coefs[3]: affine coefficients {cx, cy | cc, 0} for w0, w1, zinv.
//             The float4 layout doubles as the per-half WMMA B operand:
//             lanes 0-15 load the first 8B {cx,cy} (K0,K1), lanes 16-31 the
//             second 8B {cc,0} (K2,K3) -- pure address selection, no selects.
//   fdat[3] : projected vertices {x,y,z, shade} for perspective-correct shading
// ---------------------------------------------------------------------------
__global__ void face_prep_kernel(const float* __restrict__ verts,
                                 const int*   __restrict__ faces,
                                 float4* __restrict__ coefs,
                                 float4* __restrict__ fdat) {
    int idx = blockIdx.x * blockDim.x + threadIdx.x;
    if (idx >= BATCH * F2) return;
    int b = idx / F2, f = idx % F2;
    int fsrc = (f < NFACE) ? f : (f - NFACE);
    const int* fp = faces + ((size_t)b * NFACE + fsrc) * 3;
    int i0, i1, i2;
    if (f < NFACE) { i0 = fp[0]; i1 = fp[1]; i2 = fp[2]; }
    else           { i0 = fp[2]; i1 = fp[1]; i2 = fp[0]; }   // reversed winding

    const float* vb = verts + (size_t)b * NVERT * 3;
    float w0x = vb[i0*3+0], w0y = vb[i0*3+1], w0z = vb[i0*3+2];
    float w1x = vb[i1*3+0], w1y = vb[i1*3+1], w1z = vb[i1*3+2];
    float w2x = vb[i2*3+0], w2y = vb[i2*3+1], w2z = vb[i2*3+2];

    // lighting from world-space normal; LIGHT_DIR = (0,1,1) un-normalized
    float e1x = w1x - w0x, e1y = w1y - w0y, e1z = w1z - w0z;
    float e2x = w2x - w0x, e2y = w2y - w0y, e2z = w2z - w0z;
    float nx = e1y * e2z - e1z * e2y;
    float ny = e1z * e2x - e1x * e2z;
    float nz = e1x * e2y - e1y * e2x;
    float nl = sqrtf(nx*nx + ny*ny + nz*nz) + 1e-12f;
    float cosv = fmaxf((ny + nz) / nl, 0.0f);
    float shade = 0.5f + 0.5f * cosv;

    // look_at collapses to v - eye (R == I); then perspective
    float z0 = w0z + EYE_OFF, z1 = w1z + EYE_OFF, z2 = w2z + EYE_OFF;
    float x0 = w0x / (z0 * PW), y0 = w0y / (z0 * PW);
    float x1 = w1x / (z1 * PW), y1 = w1y / (z1 * PW);
    float x2 = w2x / (z2 * PW), y2 = w2y / (z2 * PW);

    float det = (y1 - y2) * (x0 - x2) + (x2 - x1) * (y0 - y2);
    bool ok = fabsf(det) > EPSD;
    float inv = ok ? (1.0f / det) : 1.0f;
    float a0x = (y1 - y2) * inv, a0y = (x2 - x1) * inv;
    float a0c = -(a0x * x2 + a0y * y2);
    float a1x = (y2 - y0) * inv, a1y = (x0 - x2) * inv;
    float a1c = -(a1x * x2 + a1y * y2);
    float iz0 = 1.0f / z0, iz1 = 1.0f / z1, iz2 = 1.0f / z2;
    float d0 = iz0 - iz2, d1 = iz1 - iz2;
    float zx = a0x * d0 + a1x * d1;
    float zy = a0y * d0 + a1y * d1;
    float zc = a0c * d0 + a1c * d1 + iz2;
    if (!ok) {  // degenerate face: force w0 = -1 so it never passes the hit test
        a0x = 0.0f; a0y = 0.0f; a0c = -1.0f;
        a1x = 0.0f; a1y = 0.0f; a1c = -1.0f;
        zx = 0.0f; zy = 0.0f; zc = 1.0f;
    }
    coefs[(size_t)idx*3 + 0] = make_float4(a0x, a0y, a0c, 0.0f);
    coefs[(size_t)idx*3 + 1] = make_float4(a1x, a1y, a1c, 0.0f);
    coefs[(size_t)idx*3 + 2] = make_float4(zx,  zy,  zc,  0.0f);
    fdat[(size_t)idx*3 + 0] = make_float4(x0, y0, z0, shade);
    fdat[(size_t)idx*3 + 1] = make_float4(x1, y1, z1, 0.0f);
    fdat[(size_t)idx*3 + 2] = make_float4(x2, y2, z2, 0.0f);
}

// ---------------------------------------------------------------------------
// Kernel 2: WMMA z-buffer raster. One wave32 = one 16-pixel row tile.
// Per face group (16 faces): 3x V_WMMA_F32_16X16X4_F32 evaluate w0, w1, zinv
// for 16 pixels x 16 faces. B operands are fetched as per-lane b64 loads
// whose ADDRESS encodes the K-half (no cndmask selects). Depth test runs in
// INVERSE-Z domain (zp < zbest <=> zinv > zbinv, z > 0), so no divide in the
// hot loop. Loop unrolled x4 so intra-unroll load offsets become instruction
// immediates. Final shfl_xor butterfly arg-maxes zbinv across the 16 face
// columns (tie-break: smaller face id). Control flow is fully uniform ->
// EXEC all-1s at every WMMA (ISA requirement).
// ---------------------------------------------------------------------------
__global__ void raster_kernel(const float4* __restrict__ coefs,
                              int* __restrict__ fidbuf) {
    const int gtid = blockIdx.x * blockDim.x + threadIdx.x;
    const int wid  = gtid >> 5;
    const int lane = gtid & 31;
    const int half = lane >> 4;     // A/B K-split half per ISA 16x4 f32 layout
    const int l4   = lane & 15;

    const int tilesPerRow = HRES / 16;
    const int b   = wid / (HRES * tilesPerRow);
    const int rem = wid % (HRES * tilesPerRow);
    const int row = rem / tilesPerRow;
    const int tx  = rem % tilesPerRow;
    const float py = -axf(row);

    // A (16x4): lanes 0-15 -> {K0=px, K1=py}; lanes 16-31 -> {K2=1, K3=0}
    v2f a;
    a.x = half ? 1.0f : axf(tx * 16 + l4);
    a.y = half ? 0.0f : py;
    v8f cz = {};

    const float INV_NEAR = 1.0f / NEARZ;   // 10
    float zbinv[8];                        // 1/zbest; init 1/FAR
    int   fb[8];
#pragma unroll
    for (int j = 0; j < 8; ++j) { zbinv[j] = 1.0f / FARZ; fb[j] = -1; }

    // per-lane byte base: batch + face-in-group + K-half selection by address
    const char* cpb = (const char*)(coefs + (size_t)b * F2 * 3)
                    + (size_t)l4 * 48 + (size_t)half * 8;
    int face = l4;
#pragma unroll 4
    for (int g = 0; g < F2 / 16; ++g) {
        const char* cg = cpb + (size_t)g * 768;   // 16 faces * 48B per group
        // B (4x16): lanes 0-15 -> {K0=cx, K1=cy}; lanes 16-31 -> {K2=cc, K3=0}
        float2 q0 = *(const float2*)(cg +  0);    // w0 coefficients
        float2 q1 = *(const float2*)(cg + 16);    // w1 coefficients
        float2 q2 = *(const float2*)(cg + 32);    // zinv coefficients
        v2f b0; b0.x = q0.x; b0.y = q0.y;
        v2f b1; b1.x = q1.x; b1.y = q1.y;
        v2f b2; b2.x = q2.x; b2.y = q2.y;

        v8f w0v = __builtin_amdgcn_wmma_f32_16x16x4_f32(false, a, false, b0,
                                                        (short)0, cz, false, false);
        v8f w1v = __builtin_amdgcn_wmma_f32_16x16x4_f32(false, a, false, b1,
                                                        (short)0, cz, false, false);
        v8f zvv = __builtin_amdgcn_wmma_f32_16x16x4_f32(false, a, false, b2,
                                                        (short)0, cz, false, false);
#pragma unroll
        for (int j = 0; j < 8; ++j) {
            // D element: pixel M = j + 8*half, face column N = l4
            float w0 = w0v[j], w1 = w1v[j];
            float w2 = 1.0f - w0 - w1;
            float zi = zvv[j];
            // reference quirk: |zinv|<=1e-12 -> zp = 1.0  <=>  zinv_eff = 1.0
            float zeff = (fabsf(zi) > 1e-12f) ? zi : 1.0f;
            // inside test: min(w0,w1,w2) >= 0 (v_min3_num_f32)
            float mw = fminf(fminf(w0, w1), w2);
            // inverse-z depth test: zp in (NEAR, zbest)  <=>  zbinv < zeff < 10
            bool hit = (mw >= 0.0f) & (zeff > zbinv[j]) & (zeff < INV_NEAR);
            zbinv[j] = hit ? zeff : zbinv[j];
            fb[j]    = hit ? face : fb[j];
        }
        face += 16;
    }

    // arg-max of zbinv over the 16 face columns (stays within each 16-lane half)
#pragma unroll
    for (int off = 1; off < 16; off <<= 1) {
#pragma unroll
        for (int j = 0; j < 8; ++j) {
            float oz = __shfl_xor(zbinv[j], off, 32);
            int   of = __shfl_xor(fb[j], off, 32);
            bool take = (oz > zbinv[j]) || ((oz == zbinv[j]) && (of < fb[j]));
            zbinv[j] = take ? oz : zbinv[j];
            fb[j]    = take ? of : fb[j];
        }
    }
    if (l4 == 0) {   // lane 0 writes pixels M=0..7, lane 16 writes M=8..15
        int* dst = fidbuf + ((size_t)b * HRES + row) * HRES + tx * 16 + half * 8;
#pragma unroll
        for (int j = 0; j < 8; ++j) dst[j] = fb[j];
    }
}

// ---------------------------------------------------------------------------
// Kernel 3: perspective-correct shading + 2x2 mean downsample, fused.
// Handles the textures2 transpose (reversed-winding half) by index swap.
// ---------------------------------------------------------------------------
__global__ void shade_kernel(const int*    __restrict__ fidbuf,
                             const float4* __restrict__ fdat,
                             const float*  __restrict__ tex,
                             float* __restrict__ out) {
    int idx = blockIdx.x * blockDim.x + threadIdx.x;
    if (idx >= BATCH * ORES * ORES) return;
    int b = idx / (ORES * ORES);
    int p = idx % (ORES * ORES);
    int oy = p / ORES, ox = p % ORES;

    float acc0 = 0.0f, acc1 = 0.0f, acc2 = 0.0f;
#pragma unroll
    for (int s = 0; s < 4; ++s) {
        int y = oy * 2 + (s >> 1);
        int x = ox * 2 + (s & 1);
        int fid = fidbuf[((size_t)b * HRES + y) * HRES + x];
        if (fid < 0) continue;   // background contributes 0
        const float4* fd = fdat + ((size_t)b * F2 + fid) * 3;
        float4 p0 = fd[0], p1 = fd[1], p2 = fd[2];
        float px = axf(x), py = -axf(y);
        float det = (p1.y - p2.y) * (p0.x - p2.x) + (p2.x - p1.x) * (p0.y - p2.y);
        float inv = 1.0f / (fabsf(det) > EPSD ? det : 1.0f);
        float w0 = ((p1.y - p2.y) * (px - p2.x) + (p2.x - p1.x) * (py - p2.y)) * inv;
        float w1 = ((p2.y - p0.y) * (px - p2.x) + (p0.x - p2.x) * (py - p2.y)) * inv;
        float w2 = 1.0f - w0 - w1;
        float q0 = w0 / p0.z, q1 = w1 / p1.z, q2 = w2 / p2.z;
        float zp = 1.0f / (q0 + q1 + q2);
        float wc0 = q0 * zp, wc1 = q1 * zp, wc2 = q2 * zp;
        int t0 = min(max((int)floorf(wc0 * 2.0f), 0), 1);
        int t1 = min(max((int)floorf(wc1 * 2.0f), 0), 1);
        int t2 = min(max((int)floorf(wc2 * 2.0f), 0), 1);
        int fs, a0, a1, a2;
        if (fid < NFACE) { fs = fid;          a0 = t0; a1 = t1; a2 = t2; }
        else             { fs = fid - NFACE;  a0 = t2; a1 = t1; a2 = t0; } // transpose(2,1,0)
        const float* tp = tex + ((((size_t)(b * NFACE + fs) * 2 + a0) * 2 + a1) * 2 + a2) * 3;
        float sh = p0.w;
        acc0 += tp[0] * sh; acc1 += tp[1] * sh; acc2 += tp[2] * sh;
    }
    size_t plane = (size_t)ORES * ORES;
    size_t obase = (size_t)b * 3 * plane + (size_t)oy * ORES + ox;
    out[obase]             = acc0 * 0.25f;
    out[obase + plane]     = acc1 * 0.25f;
    out[obase + 2 * plane] = acc2 * 0.25f;
}

extern "C" void kernel_launch(void* const* d_in, const int* in_sizes, int n_in,
                              void* d_out, int out_size, void* d_ws, size_t ws_size,
                              hipStream_t stream) {
    const float* verts = (const float*)d_in[0];   // (2,1282,3) f32
    const float* tex   = (const float*)d_in[1];   // (2,1280,2,2,2,3) f32
    const int*   faces = (const int*)d_in[2];     // (2,1280,3) i32
    float* out = (float*)d_out;                   // (2,3,128,128) f32

    // workspace layout (all 16B aligned):
    //   [0, 245760)        coefs  : B*F2*3 float4
    //   [245760, 491520)   fdat   : B*F2*3 float4
    //   [491520, 1015808)  fidbuf : B*256*256 int
    char* ws = (char*)d_ws;
    float4* coefs = (float4*)(ws);
    float4* fdat  = (float4*)(ws + 245760);
    int*    fidb  = (int*)  (ws + 491520);

    face_prep_kernel<<<(BATCH * F2 + 255) / 256, 256, 0, stream>>>(verts, faces, coefs, fdat);

    const int nwaves = BATCH * HRES * (HRES / 16);          // 8192 row tiles
    raster_kernel<<<nwaves * 32 / 256, 256, 0, stream>>>(coefs, fidb);

    shade_kernel<<<(BATCH * ORES * ORES + 255) / 256, 256, 0, stream>>>(fidb, fdat, tex, out);
}